// HGCRU_27556510171435
// MI455X (gfx1250) — compile-verified
//
#include <hip/hip_runtime.h>
#include <hip/hip_bf16.h>

typedef __attribute__((ext_vector_type(2))) float v2f;
typedef __attribute__((ext_vector_type(8))) float v8f;

#define LDSA_STRIDE 132   // 132 mod 64 = 4 -> conflict-free column reads, keeps 8B alignment

// ---------------------------------------------------------------------------
// Utility kernels
// ---------------------------------------------------------------------------
__global__ void k_zero(float* __restrict__ p, int n) {
    int i = blockIdx.x * blockDim.x + threadIdx.x;
    if (i < n) p[i] = 0.0f;
}

// D[node] += hw[edge];  Bcnt[edge] += 1
__global__ void k_degree(const int* __restrict__ nidx, const int* __restrict__ eidx,
                         const float* __restrict__ hw,
                         float* __restrict__ D, float* __restrict__ Bcnt, int n_inc) {
    int i = blockIdx.x * blockDim.x + threadIdx.x;
    if (i >= n_inc) return;
    int e = eidx[i];
    atomicAdd(&D[nidx[i]], hw[e]);
    atomicAdd(&Bcnt[e], 1.0f);
}

__global__ void k_invert(float* __restrict__ p, int n) {
    int i = blockIdx.x * blockDim.x + threadIdx.x;
    if (i < n) { float v = p[i]; p[i] = (v > 0.0f) ? (1.0f / v) : 0.0f; }
}

// ---------------------------------------------------------------------------
// WMMA f32 GEMM:  Y[M x 128] = A[M x 128] @ B[128 x 128]
// mode 0: A = [x | state],          B = gate_W (128x128), no store bias
// mode 1: A = [x | r*state],        B = [cand_W | cand_res_W] (each 128x64),
//         columns >=64 get + cand_res_b at store.
// One wave computes one 16x16 tile via 32x v_wmma_f32_16x16x4_f32.
// Block = 256 threads (8 waves) = 16 rows x 128 cols.
// ---------------------------------------------------------------------------
__global__ __launch_bounds__(256) void k_gemm_wmma(
    const float* __restrict__ x,     // [M,64]
    const float* __restrict__ st,    // [M,64]
    const float* __restrict__ rmul,  // [M,64] (mode 1 only)
    const float* __restrict__ W0,
    const float* __restrict__ W1,
    const float* __restrict__ bias1,
    float* __restrict__ Yout,        // [M,128]
    int M, int mode) {
    __shared__ float Atile[16 * LDSA_STRIDE];

    const int  row0 = blockIdx.x * 16;
    const bool full = (row0 + 16 <= M);

    // Stage the 16x128 A tile into LDS, building concat(x, state/r*state) on the fly.
    if (full) {
        for (int t = threadIdx.x; t < 16 * 128; t += 256) {
            int rr = t >> 7;
            int cc = t & 127;
            int gr = row0 + rr;
            float v;
            if (cc < 64) {
                v = x[gr * 64 + cc];
            } else {
                v = st[gr * 64 + cc - 64];
                if (mode == 1) v *= rmul[gr * 64 + cc - 64];
            }
            Atile[rr * LDSA_STRIDE + cc] = v;
        }
    } else {
        for (int t = threadIdx.x; t < 16 * 128; t += 256) {
            int rr = t >> 7;
            int cc = t & 127;
            int gr = row0 + rr;
            float v = 0.0f;
            if (gr < M) {
                if (cc < 64) {
                    v = x[gr * 64 + cc];
                } else {
                    v = st[gr * 64 + cc - 64];
                    if (mode == 1) v *= rmul[gr * 64 + cc - 64];
                }
            }
            Atile[rr * LDSA_STRIDE + cc] = v;
        }
    }
    __syncthreads();

    const int wave = threadIdx.x >> 5;   // column tile 0..7
    const int lane = threadIdx.x & 31;
    const int half = lane >> 4;          // 0: K={k0,k0+1}, 1: K={k0+2,k0+3}
    const int l16  = lane & 15;
    const int ncol = wave * 16 + l16;

    const float* Wp; int wstride; int wcol;
    if (mode == 0) { Wp = W0; wstride = 128; wcol = ncol; }
    else if (ncol < 64) { Wp = W0; wstride = 64; wcol = ncol; }
    else { Wp = W1; wstride = 64; wcol = ncol - 64; }

    const float* arow = &Atile[l16 * LDSA_STRIDE];

    v8f c = {0.f, 0.f, 0.f, 0.f, 0.f, 0.f, 0.f, 0.f};
#pragma unroll 8
    for (int k0 = 0; k0 < 128; k0 += 4) {
        int ka = k0 + 2 * half;
        v2f a, b;
        a.x = arow[ka];
        a.y = arow[ka + 1];
        b.x = Wp[ka * wstride + wcol];
        b.y = Wp[(ka + 1) * wstride + wcol];
        c = __builtin_amdgcn_wmma_f32_16x16x4_f32(
                /*neg_a=*/false, a, /*neg_b=*/false, b,
                /*c_mod=*/(short)0, c, /*reuse_a=*/false, /*reuse_b=*/false);
    }

    const float biasv = (mode == 1 && ncol >= 64) ? bias1[ncol - 64] : 0.0f;

    // C layout: VGPR v holds row row0+v (lanes<16) and row row0+v+8 (lanes>=16).
    if (full) {
        float* outp = Yout + (size_t)(row0 + 8 * half) * 128 + ncol;
#pragma unroll
        for (int v = 0; v < 8; ++v) outp[v * 128] = c[v] + biasv;
    } else {
#pragma unroll
        for (int v = 0; v < 8; ++v) {
            int gr = row0 + v + 8 * half;
            if (gr < M) Yout[(size_t)gr * 128 + ncol] = c[v] + biasv;
        }
    }
}

// ---------------------------------------------------------------------------
// Segment sums (atomic scatter-add). One warp covers one incidence (F/4 chunks
// of 4 floats) so loads/atomics are coalesced along the feature dim.
// ---------------------------------------------------------------------------
__global__ void k_edge_agg(const int* __restrict__ nidx, const int* __restrict__ eidx,
                           const float* __restrict__ Y, int ystride,
                           float* __restrict__ E, int F, int n_inc) {
    int chunks = F >> 2;
    long long tid = (long long)blockIdx.x * blockDim.x + threadIdx.x;
    if (tid >= (long long)n_inc * chunks) return;
    int i  = (int)(tid / chunks);
    int c  = ((int)(tid % chunks)) << 2;
    int n  = nidx[i];
    int e  = eidx[i];
    const float4 s4 = *reinterpret_cast<const float4*>(Y + (long long)n * ystride + c);
    float* dst = E + (long long)e * F + c;
    atomicAdd(dst + 0, s4.x);
    atomicAdd(dst + 1, s4.y);
    atomicAdd(dst + 2, s4.z);
    atomicAdd(dst + 3, s4.w);
}

__global__ void k_edge_scale(float* __restrict__ E, const float* __restrict__ Binv,
                             int F, int n_edges) {
    int i = blockIdx.x * blockDim.x + threadIdx.x;
    if (i < n_edges * F) E[i] *= Binv[i / F];
}

__global__ void k_node_agg(const int* __restrict__ nidx, const int* __restrict__ eidx,
                           const float* __restrict__ hw, const float* __restrict__ m,
                           float* __restrict__ A, int F, int n_inc) {
    int chunks = F >> 2;
    long long tid = (long long)blockIdx.x * blockDim.x + threadIdx.x;
    if (tid >= (long long)n_inc * chunks) return;
    int i  = (int)(tid / chunks);
    int c  = ((int)(tid % chunks)) << 2;
    int n  = nidx[i];
    int e  = eidx[i];
    float w = hw[e];
    const float4 s4 = *reinterpret_cast<const float4*>(m + (long long)e * F + c);
    float* dst = A + (long long)n * F + c;
    atomicAdd(dst + 0, w * s4.x);
    atomicAdd(dst + 1, w * s4.y);
    atomicAdd(dst + 2, w * s4.z);
    atomicAdd(dst + 3, w * s4.w);
}

// ---------------------------------------------------------------------------
// Gate epilogue: z_r = sigmoid(LN(relu(xs + Dinv*A + gate_b)))   (128 features)
// One wave per row; lane holds 4 features.
// ---------------------------------------------------------------------------
__global__ __launch_bounds__(256) void k_gate_final(
    const float* __restrict__ x, const float* __restrict__ st,
    const float* __restrict__ A, const float* __restrict__ Dinv,
    const float* __restrict__ gb, const float* __restrict__ lng,
    const float* __restrict__ lnb,
    float* __restrict__ zb, float* __restrict__ rb, int M) {
    int warp = threadIdx.x >> 5;
    int lane = threadIdx.x & 31;
    int row = blockIdx.x * 8 + warp;
    if (row >= M) return;
    float di = Dinv[row];
    float h[4];
    float sum = 0.0f;
#pragma unroll
    for (int j = 0; j < 4; ++j) {
        int c = lane * 4 + j;
        float xs = (c < 64) ? x[row * 64 + c] : st[row * 64 + c - 64];
        float v = xs + di * A[row * 128 + c] + gb[c];
        v = fmaxf(v, 0.0f);
        h[j] = v;
        sum += v;
    }
#pragma unroll
    for (int off = 16; off > 0; off >>= 1) sum += __shfl_xor(sum, off, 32);
    float mu = sum * (1.0f / 128.0f);
    float var = 0.0f;
#pragma unroll
    for (int j = 0; j < 4; ++j) { float d = h[j] - mu; var += d * d; }
#pragma unroll
    for (int off = 16; off > 0; off >>= 1) var += __shfl_xor(var, off, 32);
    var *= (1.0f / 128.0f);
    float rs = rsqrtf(var + 1e-5f);
#pragma unroll
    for (int j = 0; j < 4; ++j) {
        int c = lane * 4 + j;
        float y = (h[j] - mu) * rs * lng[c] + lnb[c];
        float sg = 1.0f / (1.0f + __expf(-y));
        if (c < 64) zb[row * 64 + c] = sg;
        else        rb[row * 64 + c - 64] = sg;
    }
}

// ---------------------------------------------------------------------------
// Candidate epilogue: hc = tanh(LN(relu(resid + Dinv*A2 + cand_b)))  (64 feats)
// out = (1-z)*state + z*hc
// ---------------------------------------------------------------------------
__global__ __launch_bounds__(256) void k_cand_final(
    const float* __restrict__ Ycat,   // [M,128]; cols 64..127 = residual (+cand_res_b)
    const float* __restrict__ A2,     // [M,64]
    const float* __restrict__ Dinv,
    const float* __restrict__ cb, const float* __restrict__ lng,
    const float* __restrict__ lnb,
    const float* __restrict__ zb, const float* __restrict__ st,
    float* __restrict__ out, int M) {
    int warp = threadIdx.x >> 5;
    int lane = threadIdx.x & 31;
    int row = blockIdx.x * 8 + warp;
    if (row >= M) return;
    float di = Dinv[row];
    float h[2];
    float sum = 0.0f;
#pragma unroll
    for (int j = 0; j < 2; ++j) {
        int c = lane * 2 + j;
        float resid = Ycat[row * 128 + 64 + c];
        float v = resid + di * A2[row * 64 + c] + cb[c];
        v = fmaxf(v, 0.0f);
        h[j] = v;
        sum += v;
    }
#pragma unroll
    for (int off = 16; off > 0; off >>= 1) sum += __shfl_xor(sum, off, 32);
    float mu = sum * (1.0f / 64.0f);
    float var = 0.0f;
#pragma unroll
    for (int j = 0; j < 2; ++j) { float d = h[j] - mu; var += d * d; }
#pragma unroll
    for (int off = 16; off > 0; off >>= 1) var += __shfl_xor(var, off, 32);
    var *= (1.0f / 64.0f);
    float rs = rsqrtf(var + 1e-5f);
#pragma unroll
    for (int j = 0; j < 2; ++j) {
        int c = lane * 2 + j;
        float y = (h[j] - mu) * rs * lng[c] + lnb[c];
        float hc = tanhf(y);
        float z = zb[row * 64 + c];
        out[row * 64 + c] = (1.0f - z) * st[row * 64 + c] + z * hc;
    }
}

// ---------------------------------------------------------------------------
// Launch
// ---------------------------------------------------------------------------
extern "C" void kernel_launch(void* const* d_in, const int* in_sizes, int n_in,
                              void* d_out, int out_size, void* d_ws, size_t ws_size,
                              hipStream_t stream) {
    const float* x    = (const float*)d_in[0];
    const float* st   = (const float*)d_in[1];
    const int*   hix  = (const int*)d_in[2];
    const float* hw   = (const float*)d_in[3];
    const float* gW   = (const float*)d_in[4];
    const float* gb   = (const float*)d_in[5];
    const float* glng = (const float*)d_in[6];
    const float* glnb = (const float*)d_in[7];
    const float* cW   = (const float*)d_in[8];
    const float* cb   = (const float*)d_in[9];
    const float* clng = (const float*)d_in[10];
    const float* clnb = (const float*)d_in[11];
    const float* crW  = (const float*)d_in[12];
    const float* crb  = (const float*)d_in[13];
    float* out = (float*)d_out;

    const int nN = in_sizes[0] / 64;   // 50000 nodes
    const int nI = in_sizes[2] / 2;    // 800000 incidences
    const int nE = in_sizes[3];        // 10000 edges
    const int* nidx = hix;
    const int* eidx = hix + nI;

    float* ws  = (float*)d_ws;
    float* Dv  = ws;                       // nN       (degree -> Dinv)
    float* Bv  = Dv + nN;                  // nE       (count  -> Binv)
    float* Y   = Bv + nE;                  // nN*128   (GEMM out, reused)
    float* Eb  = Y + (size_t)nN * 128;     // nE*128   (edge accum, reused)
    float* Ab  = Eb + (size_t)nE * 128;    // nN*128   (node accum, reused)
    float* zb  = Ab + (size_t)nN * 128;    // nN*64
    float* rb  = zb + (size_t)nN * 64;     // nN*64

    const int T = 256;
    auto blocks = [](long long n, int t) { return (unsigned)((n + t - 1) / t); };

    // ---- degrees / counts (shared by both blocks) ----
    k_zero<<<blocks(nN, T), T, 0, stream>>>(Dv, nN);
    k_zero<<<blocks(nE, T), T, 0, stream>>>(Bv, nE);
    k_degree<<<blocks(nI, T), T, 0, stream>>>(nidx, eidx, hw, Dv, Bv, nI);
    k_invert<<<blocks(nN, T), T, 0, stream>>>(Dv, nN);
    k_invert<<<blocks(nE, T), T, 0, stream>>>(Bv, nE);

    // ---- gate path (F = 128) ----
    k_zero<<<blocks((long long)nE * 128, T), T, 0, stream>>>(Eb, nE * 128);
    k_zero<<<blocks((long long)nN * 128, T), T, 0, stream>>>(Ab, nN * 128);
    k_gemm_wmma<<<(nN + 15) / 16, 256, 0, stream>>>(x, st, x, gW, gW, gb, Y, nN, 0);
    k_edge_agg<<<blocks((long long)nI * 32, T), T, 0, stream>>>(nidx, eidx, Y, 128, Eb, 128, nI);
    k_edge_scale<<<blocks((long long)nE * 128, T), T, 0, stream>>>(Eb, Bv, 128, nE);
    k_node_agg<<<blocks((long long)nI * 32, T), T, 0, stream>>>(nidx, eidx, hw, Eb, Ab, 128, nI);
    k_gate_final<<<(nN + 7) / 8, 256, 0, stream>>>(x, st, Ab, Dv, gb, glng, glnb, zb, rb, nN);

    // ---- candidate path (F = 64), reuse Y/Eb/Ab ----
    k_zero<<<blocks((long long)nE * 64, T), T, 0, stream>>>(Eb, nE * 64);
    k_zero<<<blocks((long long)nN * 64, T), T, 0, stream>>>(Ab, nN * 64);
    k_gemm_wmma<<<(nN + 15) / 16, 256, 0, stream>>>(x, st, rb, cW, crW, crb, Y, nN, 1);
    k_edge_agg<<<blocks((long long)nI * 16, T), T, 0, stream>>>(nidx, eidx, Y, 128, Eb, 64, nI);
    k_edge_scale<<<blocks((long long)nE * 64, T), T, 0, stream>>>(Eb, Bv, 64, nE);
    k_node_agg<<<blocks((long long)nI * 16, T), T, 0, stream>>>(nidx, eidx, hw, Eb, Ab, 64, nI);
    k_cand_final<<<(nN + 7) / 8, 256, 0, stream>>>(Y, Ab, Dv, cb, clng, clnb, zb, st, out, nN);
}